// MultiHeadAttention_10299331575855
// MI455X (gfx1250) — compile-verified
//
#include <hip/hip_runtime.h>

// MI455X / gfx1250, wave32. Matrix ops via V_WMMA_F32_16X16X4_F32.
typedef __attribute__((ext_vector_type(2))) float v2f;
typedef __attribute__((ext_vector_type(8))) float v8f;
typedef __attribute__((ext_vector_type(4))) int v4i;

#define AS1 __attribute__((address_space(1)))
#define AS3 __attribute__((address_space(3)))

#if defined(__has_builtin)
#if __has_builtin(__builtin_amdgcn_global_load_async_to_lds_b128) && \
    __has_builtin(__builtin_amdgcn_s_wait_asynccnt)
#define USE_ASYNC_LDS 1
#endif
#endif
#ifndef USE_ASYNC_LDS
#define USE_ASYNC_LDS 0
#endif

// copy 16B global -> LDS (async on CDNA5, sync fallback otherwise)
static __device__ __forceinline__ void cp16(const float* __restrict__ g,
                                            float* __restrict__ l) {
#if USE_ASYNC_LDS
  __builtin_amdgcn_global_load_async_to_lds_b128((AS1 v4i*)(g), (AS3 v4i*)(l),
                                                 0, 0);
#else
  *(float4*)l = *(const float4*)g;
#endif
}
static __device__ __forceinline__ void cp_wait0() {
#if USE_ASYNC_LDS
  __builtin_amdgcn_s_wait_asynccnt(0);
#endif
}
static __device__ __forceinline__ void cp_wait8() {
#if USE_ASYNC_LDS
  __builtin_amdgcn_s_wait_asynccnt(8);
#endif
}

static __device__ __forceinline__ v8f wmma4(v2f a, v2f b, v8f c) {
  // D(16x16,f32) = A(16x4,f32) * B(4x16,f32) + C
  return __builtin_amdgcn_wmma_f32_16x16x4_f32(
      false, a, false, b, (short)0, c, false, false);
}

// maintain sorted-descending top-5 (a0 >= a1 >= ... >= a4)
static __device__ __forceinline__ void ins5(float vv, float& a0, float& a1,
                                            float& a2, float& a3, float& a4) {
  if (vv <= a4) return;
  a4 = vv;
  if (a4 > a3) { float t = a3; a3 = a4; a4 = t; }
  if (a3 > a2) { float t = a2; a2 = a3; a3 = t; }
  if (a2 > a1) { float t = a1; a1 = a2; a2 = t; }
  if (a1 > a0) { float t = a0; a0 = a1; a1 = t; }
}

// ---------------------------------------------------------------------------
// Kernel 1: qkv = x(4096x512) @ w_qkv(512x1536) + b_qkv, scattered to
// head-major q/k/v [B,H,S,64]. Block = 128 thr (4 waves) = 16 rows x 64 cols.
// Both X (16x64) and W (64x64) tiles async-staged in LDS per K-chunk.
// ---------------------------------------------------------------------------
__global__ __launch_bounds__(128) void qkv_kernel(
    const float* __restrict__ x, const float* __restrict__ w,
    const float* __restrict__ bias, float* __restrict__ qo,
    float* __restrict__ ko, float* __restrict__ vo) {
  __shared__ float sX[16][68];   // 68 % 64 = 4 -> conflict-free A frags
  __shared__ float sW[64][80];   // 160 % 64 = 32 -> hi/lo half disjoint banks
  const int tid  = threadIdx.x;
  const int lane = tid & 31;
  const int wv   = tid >> 5;
  const int half = lane >> 4;    // 0: K 0,1   1: K 2,3
  const int koff = half * 2;
  const int rl   = lane & 15;
  const int r0   = blockIdx.x * 16;
  const int c0b  = blockIdx.y * 64;          // block column base
  const int cw   = wv * 16;                  // wave column offset in block
  v8f acc = {};
  for (int k0 = 0; k0 < 512; k0 += 64) {
    for (int j = 0; j < 2; ++j) {            // X tile: 256 b128 / 128 thr
      int idx = j * 128 + tid;
      int rr = idx >> 4, dd = (idx & 15) * 4;
      cp16(x + (size_t)(r0 + rr) * 512 + k0 + dd, &sX[rr][dd]);
    }
    for (int j = 0; j < 8; ++j) {            // W tile: 1024 b128 / 128 thr
      int idx = j * 128 + tid;
      int rr = idx >> 4, cc = (idx & 15) * 4;
      cp16(w + (size_t)(k0 + rr) * 1536 + c0b + cc, &sW[rr][cc]);
    }
    cp_wait0();
    __syncthreads();
#pragma unroll
    for (int d0 = 0; d0 < 64; d0 += 4) {
      v2f a = *(const v2f*)&sX[rl][d0 + koff];   // A: row rl, K pair
      v2f b;
      b.x = sW[d0 + koff][cw + rl];              // B[K][N]
      b.y = sW[d0 + koff + 1][cw + rl];
      acc = wmma4(a, b, acc);
    }
    __syncthreads();
  }
#pragma unroll
  for (int j = 0; j < 8; ++j) {   // C/D: vgpr j -> row j (+8 for hi half)
    int row = j + half * 8;
    int gr = r0 + row, gc = c0b + cw + rl;
    float val = acc[j] + bias[gc];
    int which = gc >> 9;          // 0=q 1=k 2=v
    int d = gc & 511, h = d >> 6, hd = d & 63;
    int bb = gr >> 11, s = gr & 2047;
    float* dst = (which == 0) ? qo : ((which == 1) ? ko : vo);
    dst[((size_t)(bb * 8 + h) * 2048 + s) * 64 + hd] = val;
  }
}

// ---------------------------------------------------------------------------
// Kernel 2: feat[b,h,s] = sigmoid(mean_d q[b,h,s,d]); one wave32 per row.
// ---------------------------------------------------------------------------
__global__ __launch_bounds__(256) void feat_kernel(const float* __restrict__ q,
                                                   float* __restrict__ feat) {
  int row  = blockIdx.x * 8 + (threadIdx.x >> 5);
  int lane = threadIdx.x & 31;
  float s = q[(size_t)row * 64 + lane] + q[(size_t)row * 64 + 32 + lane];
  for (int m = 16; m; m >>= 1) s += __shfl_xor(s, m, 32);
  if (lane == 0) feat[row] = 1.0f / (1.0f + __expf(-s * (1.0f / 64.0f)));
}

// ---------------------------------------------------------------------------
// Kernel 3: fused sparse attention for one (b, h, 16-query tile).
// 4 waves: wave w owns score columns [w*512, w*512+512); K tiles are
// double-buffered with async-to-LDS copies overlapped with WMMA.
// ---------------------------------------------------------------------------
__global__ __launch_bounds__(128) void attn_kernel(
    const float* __restrict__ q, const float* __restrict__ k,
    const float* __restrict__ v, const float* __restrict__ feat,
    float* __restrict__ attn_out, float* __restrict__ ctx) {
  __shared__ float sS[16][2052];       // 16 x 2048 scores (padded stride)
  __shared__ float sQ[16][68];
  __shared__ float sK[4][2][16][68];   // per-wave double-buffered K staging
  __shared__ float sInv[16];
  const int tid  = threadIdx.x;
  const int lane = tid & 31;
  const int wv   = tid >> 5;
  const int half = lane >> 4;
  const int koff = half * 2;
  const int rl   = lane & 15;
  const int q0 = blockIdx.x * 16;
  const int h = blockIdx.y, bb = blockIdx.z;
  const int bh = bb * 8 + h;
  const float* qg = q + (size_t)bh * 2048 * 64;
  const float* kg = k + (size_t)bh * 2048 * 64;
  const float* vg = v + (size_t)bh * 2048 * 64;
  const float* fg = feat + (size_t)bh * 2048;

  // issue Q tile (2 async/thread), then K tile 0 (8 async/wave-lane)
  for (int j = 0; j < 2; ++j) {
    int idx = j * 128 + tid;
    int rr = idx >> 4, dd = (idx & 15) * 4;
    cp16(qg + (size_t)(q0 + rr) * 64 + dd, &sQ[rr][dd]);
  }
  {
    int cbase = wv * 512;
    for (int j = 0; j < 8; ++j) {
      int idx = j * 32 + lane;
      int rr = idx >> 4, dd = (idx & 15) * 4;
      cp16(kg + (size_t)(cbase + rr) * 64 + dd, &sK[wv][0][rr][dd]);
    }
  }
  cp_wait8();        // async loads retire in order: Q's 2 are done
  __syncthreads();   // share Q across waves

  // ---- scores = (Q K^T)/8 : 16x16 WMMA tiles, K double-buffered ----------
  for (int ct = 0; ct < 32; ++ct) {
    int buf = ct & 1;
    if (ct < 31) {   // prefetch next K tile into other buffer
      int cbase = wv * 512 + (ct + 1) * 16;
      for (int j = 0; j < 8; ++j) {
        int idx = j * 32 + lane;
        int rr = idx >> 4, dd = (idx & 15) * 4;
        cp16(kg + (size_t)(cbase + rr) * 64 + dd, &sK[wv][buf ^ 1][rr][dd]);
      }
      cp_wait8();    // current tile's 8 loads complete; prefetch in flight
    } else {
      cp_wait0();
    }
    int cbase = wv * 512 + ct * 16;
    v8f acc = {};
#pragma unroll
    for (int d0 = 0; d0 < 64; d0 += 4) {
      v2f a  = *(const v2f*)&sQ[rl][d0 + koff];
      v2f b2 = *(const v2f*)&sK[wv][buf][rl][d0 + koff];  // B[K][N]
      acc = wmma4(a, b2, acc);
    }
#pragma unroll
    for (int j = 0; j < 8; ++j)
      sS[j + half * 8][cbase + rl] = acc[j] * 0.125f;
  }
  __syncthreads();

  // ---- per-row sparse mask + softmax (wave wv owns rows wv, wv+4, ...) ----
  for (int ri = 0; ri < 4; ++ri) {
    int r = wv + ri * 4;
    int rg = q0 + r;
    // global top-5 threshold
    float t0 = -3.0e38f, t1 = t0, t2 = t0, t3 = t0, t4 = t0;
    for (int i = 0; i < 64; ++i) ins5(sS[r][i * 32 + lane], t0, t1, t2, t3, t4);
    for (int m = 16; m; m >>= 1) {
      float o0 = __shfl_xor(t0, m, 32), o1 = __shfl_xor(t1, m, 32);
      float o2 = __shfl_xor(t2, m, 32), o3 = __shfl_xor(t3, m, 32);
      float o4 = __shfl_xor(t4, m, 32);
      ins5(o0, t0, t1, t2, t3, t4); ins5(o1, t0, t1, t2, t3, t4);
      ins5(o2, t0, t1, t2, t3, t4); ins5(o3, t0, t1, t2, t3, t4);
      ins5(o4, t0, t1, t2, t3, t4);
    }
    float thr_g = t4;
    // local band (covered cols lie in [rg-3, rg+3]); snapshot before rewrite
    bool okband = true;
    int bc = rg - 3 + lane;
    if (lane < 7 && bc >= 0 && bc < 2048) {
      int mn = bc < rg ? bc : rg, mx = bc < rg ? rg : bc;
      int own = mn & ~1; if (own > 2044) own = 2044;
      if (own >= mx - 3) {
        float x0 = sS[r][own], x1 = sS[r][own + 1];
        float x2 = sS[r][own + 2], x3 = sS[r][own + 3];
        float mn1 = fminf(x0, x1), mx1 = fmaxf(x0, x1);
        float mn2 = fminf(x2, x3), mx2 = fmaxf(x2, x3);
        float second = fminf(fmaxf(mn1, mn2), fminf(mx1, mx2)); // 3rd largest
        okband = sS[r][bc] >= second;
      }
    }
    unsigned bandbits = (unsigned)__ballot(okband);
    // masked rewrite + row max
    float rmax = -3.0e38f;
    for (int i = 0; i < 64; ++i) {
      int c = i * 32 + lane;
      float sv = sS[r][c];
      bool keep = sv >= thr_g;
      if (c >= rg - 3 && c <= rg + 3) {
        int mn = c < rg ? c : rg, mx = c < rg ? rg : c;
        int own = mn & ~1; if (own > 2044) own = 2044;
        if (own >= mx - 3) keep = keep && ((bandbits >> (c - (rg - 3))) & 1u);
      }
      float s2 = keep ? sv : -1.0e9f;
      rmax = fmaxf(rmax, s2);
      sS[r][c] = s2;
    }
    for (int m = 16; m; m >>= 1) rmax = fmaxf(rmax, __shfl_xor(rmax, m, 32));
    float rsum = 0.0f;
    for (int i = 0; i < 64; ++i) {
      int c = i * 32 + lane;
      float e = __expf(sS[r][c] - rmax);
      rsum += e;
      sS[r][c] = e;                    // keep unnormalized exp for attn@V
    }
    for (int m = 16; m; m >>= 1) rsum += __shfl_xor(rsum, m, 32);
    float inv = 1.0f / rsum;
    if (lane == 0) sInv[r] = inv;
    float* arow = attn_out + ((size_t)bh * 2048 + rg) * 2048;
    for (int i = 0; i < 64; ++i) {     // dense attn: streaming NT stores
      int c = i * 32 + lane;
      __builtin_nontemporal_store(sS[r][c] * inv, &arow[c]);
    }
  }
  __syncthreads();

  // ---- out = attn @ (v * feat), wave wv owns v-columns [wv*16, wv*16+16) --
  int vcol = wv * 16;
  v8f o8 = {};
  for (int k0 = 0; k0 < 2048; k0 += 4) {
    v2f a = *(const v2f*)&sS[rl][k0 + koff];
    int kr = k0 + koff;
    float f0 = fg[kr], f1 = fg[kr + 1];
    v2f b2;
    b2.x = vg[(size_t)kr * 64 + vcol + rl] * f0;
    b2.y = vg[(size_t)(kr + 1) * 64 + vcol + rl] * f1;
    o8 = wmma4(a, b2, o8);
  }
#pragma unroll
  for (int j = 0; j < 8; ++j) {
    int row = j + half * 8;
    int gq = q0 + row;
    ctx[((size_t)bb * 2048 + gq) * 512 + h * 64 + vcol + rl] =
        o8[j] * sInv[row];
  }
}

// ---------------------------------------------------------------------------
// Kernel 4: out = ctx(4096x512) @ w_proj(512x512) + b_proj
// ---------------------------------------------------------------------------
__global__ __launch_bounds__(128) void proj_kernel(
    const float* __restrict__ ctx, const float* __restrict__ w,
    const float* __restrict__ bias, float* __restrict__ out) {
  __shared__ float sX[16][68];
  __shared__ float sW[64][80];
  const int tid = threadIdx.x;
  const int lane = tid & 31;
  const int wv = tid >> 5;
  const int half = lane >> 4;
  const int koff = half * 2;
  const int rl = lane & 15;
  const int r0 = blockIdx.x * 16;
  const int c0b = blockIdx.y * 64;
  const int cw = wv * 16;
  v8f acc = {};
  for (int k0 = 0; k0 < 512; k0 += 64) {
    for (int j = 0; j < 2; ++j) {
      int idx = j * 128 + tid;
      int rr = idx >> 4, dd = (idx & 15) * 4;
      cp16(ctx + (size_t)(r0 + rr) * 512 + k0 + dd, &sX[rr][dd]);
    }
    for (int j = 0; j < 8; ++j) {
      int idx = j * 128 + tid;
      int rr = idx >> 4, cc = (idx & 15) * 4;
      cp16(w + (size_t)(k0 + rr) * 512 + c0b + cc, &sW[rr][cc]);
    }
    cp_wait0();
    __syncthreads();
#pragma unroll
    for (int d0 = 0; d0 < 64; d0 += 4) {
      v2f a = *(const v2f*)&sX[rl][d0 + koff];
      v2f b;
      b.x = sW[d0 + koff][cw + rl];
      b.y = sW[d0 + koff + 1][cw + rl];
      acc = wmma4(a, b, acc);
    }
    __syncthreads();
  }
#pragma unroll
  for (int j = 0; j < 8; ++j) {
    int row = j + half * 8;
    int gr = r0 + row, gc = c0b + cw + rl;
    out[(size_t)gr * 512 + gc] = acc[j] + bias[gc];
  }
}

// ---------------------------------------------------------------------------
extern "C" void kernel_launch(void* const* d_in, const int* in_sizes, int n_in,
                              void* d_out, int out_size, void* d_ws,
                              size_t ws_size, hipStream_t stream) {
  const float* x      = (const float*)d_in[0];
  const float* w_qkv  = (const float*)d_in[1];
  const float* b_qkv  = (const float*)d_in[2];
  const float* w_proj = (const float*)d_in[3];
  const float* b_proj = (const float*)d_in[4];
  float* out  = (float*)d_out;                    // [B,S,D] = 2,097,152 f32
  float* attn = out + (size_t)2 * 2048 * 512;     // [B,H,S,S] follows

  float* ws    = (float*)d_ws;                    // ~33.7 MB used
  float* qw    = ws;                              // [B,H,S,64]
  float* kw    = qw + (size_t)2097152;
  float* vw    = kw + (size_t)2097152;
  float* featw = vw + (size_t)2097152;            // [B,H,S]
  float* ctxw  = featw + (size_t)32768;           // [B,S,D]

  qkv_kernel<<<dim3(256, 24), 128, 0, stream>>>(x, w_qkv, b_qkv, qw, kw, vw);
  feat_kernel<<<4096, 256, 0, stream>>>(qw, featw);
  attn_kernel<<<dim3(128, 8, 2), 128, 0, stream>>>(qw, kw, vw, featw, attn,
                                                   ctxw);
  proj_kernel<<<dim3(256, 8), 128, 0, stream>>>(ctxw, w_proj, b_proj, out);
}